// HeteroDotProductPredictor_49323404427315
// MI455X (gfx1250) — compile-verified
//
#include <hip/hip_runtime.h>

// CDNA5 / gfx1250: per-edge dot product via V_WMMA_F32_16X16X4_F32.
// score[e] = dot(h[src[e]], h[dst[e]]), D_FEAT = 128, fp32 throughout.

typedef float v2f __attribute__((ext_vector_type(2)));
typedef float v8f __attribute__((ext_vector_type(8)));

#define D_FEAT 128
#define EDGES_PER_WAVE 16
#define WAVES_PER_BLOCK 8

__global__ __launch_bounds__(256) void edge_dot_wmma_f32(
    const float* __restrict__ h,
    const int* __restrict__ src,
    const int* __restrict__ dst,
    float* __restrict__ out,
    int n_edges)
{
    const int lane = threadIdx.x & 31;
    const int wave = threadIdx.x >> 5;
    const int tile = blockIdx.x * WAVES_PER_BLOCK + wave;  // 16 edges per tile
    const int e0 = tile * EDGES_PER_WAVE;
    if (e0 >= n_edges) return;  // wave-uniform: EXEC stays all-ones for WMMA

    // Edge served by this lane (lanes L and L+16 serve the same edge, different K-halves)
    const int m = lane & 15;
    const int e = min(e0 + m, n_edges - 1);

    const long long srow = (long long)src[e] * D_FEAT;
    const long long drow = (long long)dst[e] * D_FEAT;

    // A-fragment (16x4 f32, ISA 7.12.2): lane<16 -> K={0,1}, lane>=16 -> K={2,3}.
    // B-fragment (4x16) mirrors A's layout under transpose, so dst rows are
    // gathered with the identical per-lane addressing.
    const int colBase = (lane >> 4) << 1;  // 0 for lanes 0-15, 2 for lanes 16-31

    const float* __restrict__ sa = h + srow + colBase;
    const float* __restrict__ sb = h + drow + colBase;

    v8f c = {0.f, 0.f, 0.f, 0.f, 0.f, 0.f, 0.f, 0.f};

#pragma unroll
    for (int k = 0; k < D_FEAT; k += 4) {
        v2f a = *(const v2f*)(sa + k);
        v2f b = *(const v2f*)(sb + k);
        // 8 args: (neg_a, A, neg_b, B, c_mod, C, reuse_a, reuse_b)
        c = __builtin_amdgcn_wmma_f32_16x16x4_f32(
                false, a, false, b, (short)0, c, false, false);
    }

    // Extract diag(D). 16x16 f32 C/D layout: VGPR r holds M=r (lanes 0-15) and
    // M=r+8 (lanes 16-31), N = lane%16.
    //   D[m][m], m<8  -> c[m]   on lane m
    //   D[m][m], m>=8 -> c[m-8] on lane m+16  (i.e. lanes 24..31)
    float val = 0.f;
#pragma unroll
    for (int r = 0; r < 8; ++r) {
        val = (lane == r)      ? c[r] : val;
        val = (lane == r + 24) ? c[r] : val;
    }

    int eo = -1;
    if (lane < 8)        eo = e0 + lane;         // edges 0..7 of tile
    else if (lane >= 24) eo = e0 + (lane - 16);  // edges 8..15 of tile
    if (eo >= 0 && eo < n_edges) out[eo] = val;
}

extern "C" void kernel_launch(void* const* d_in, const int* in_sizes, int n_in,
                              void* d_out, int out_size, void* d_ws, size_t ws_size,
                              hipStream_t stream) {
    const float* h   = (const float*)d_in[0];
    const int*   src = (const int*)d_in[1];
    const int*   dst = (const int*)d_in[2];
    float*       out = (float*)d_out;

    const int n_edges = in_sizes[1];
    const int n_tiles = (n_edges + EDGES_PER_WAVE - 1) / EDGES_PER_WAVE;
    const int blocks  = (n_tiles + WAVES_PER_BLOCK - 1) / WAVES_PER_BLOCK;

    edge_dot_wmma_f32<<<blocks, 256, 0, stream>>>(h, src, dst, out, n_edges);
}